// MorphogenesisDynamics_15814069584137
// MI455X (gfx1250) — compile-verified
//
#include <hip/hip_runtime.h>

typedef float v2f __attribute__((ext_vector_type(2)));
typedef float v4f __attribute__((ext_vector_type(4)));
typedef float v8f __attribute__((ext_vector_type(8)));
typedef unsigned int v4u __attribute__((ext_vector_type(4)));
typedef int v4i __attribute__((ext_vector_type(4)));
typedef int v8i __attribute__((ext_vector_type(8)));

static constexpr int Hdim = 1024;
static constexpr int Edim = 8;
static constexpr int TOK = 16;        // tokens per workgroup
static constexpr int NWAVE = 4;       // waves per workgroup (128 threads)
static constexpr int ROWSTRIDE = 1032; // padded LDS row stride (floats) to avoid bank conflicts

__global__ __launch_bounds__(128)
void fused_catastrophe_fermentation(
    const float* __restrict__ x,       const float* __restrict__ control,
    const float* __restrict__ Wc,      const float* __restrict__ bc,
    const float* __restrict__ Wfc,     const float* __restrict__ bfc,
    const float* __restrict__ g1,      const float* __restrict__ b1,
    const float* __restrict__ We,      const float* __restrict__ be,
    const float* __restrict__ Vmax,    const float* __restrict__ Km,
    const float* __restrict__ Winh,    const float* __restrict__ growth_rate_p,
    const float* __restrict__ Wo,      const float* __restrict__ bo,
    const float* __restrict__ g2,      const float* __restrict__ b2,
    float* __restrict__ out)
{
    __shared__ float xbuf[TOK * ROWSTRIDE];          // 66048 B: x tile -> pre-LN1 -> pre-LN2 (in place)
    __shared__ float cscratch[NWAVE][16][16];        // 4 KB: partial C reduction for GEMM1/GEMM3
    __shared__ float eqbuf[TOK][Edim];               // cusp equilibria
    __shared__ float subbuf[TOK][Edim];              // substrate
    __shared__ float prodbuf[TOK][Edim];             // product
    __shared__ float stats[TOK][4];                  // mean1, rstd1, mean2, rstd2

    const int tid   = threadIdx.x;
    const int wave  = tid >> 5;
    const int lane  = tid & 31;
    const int lhalf = lane >> 4;       // 0: lanes 0-15, 1: lanes 16-31
    const int l16   = lane & 15;
    const int koff  = lhalf * 2;       // K offset inside a 16x16x4 fragment
    const int tokbase = blockIdx.x * TOK;

    const int eIdx = (l16 < Edim) ? l16 : (Edim - 1);   // clamped enzyme index (cols 8..15 unused)

    // ---------------- Phase 0: TDM-stage x tile into LDS ----------------
    // One tensor_load_to_lds per workgroup (issued by wave 0; EXEC-independent DMA).
    // D# uses iterate mode: tile = one 1024-float row; 16 iterations advancing
    // global by 1024 elements and LDS by 1032 elements -> padded ROWSTRIDE layout.
    if (wave == 0) {
        unsigned long long gaddr = (unsigned long long)(const void*)(x + (size_t)tokbase * Hdim);
        unsigned int lds_addr = (unsigned int)(unsigned long long)(const void*)&xbuf[0];

        v4u g0;
        g0[0] = 1u;                                   // count=1, is_restore=0, gather=0
        g0[1] = lds_addr;                             // lds_addr (bytes)
        g0[2] = (unsigned int)(gaddr & 0xFFFFFFFFull);        // global_addr[31:0]
        g0[3] = (unsigned int)((gaddr >> 32) & 0x1FFFFFFull)  // global_addr[56:32]
                | 0x80000000u;                        // type=2 ("image") in bits [127:126]

        v8i g1d;
        g1d[0] = (2 << 16)        // data_size = 4 bytes
               | (1 << 19);       // iterate_enable = 1 (no pad, no atomic barrier, mask=0)
        g1d[1] = (Hdim & 0xFFFF) << 16;               // tensor_dim0[15:0] = 1024
        g1d[2] = (1 << 16);                           // tensor_dim0[31:16]=0 | tensor_dim1[15:0]=1
        g1d[3] = (Hdim & 0xFFFF) << 16;               // tensor_dim1[31:16]=0 | tile_dim0 = 1024
        g1d[4] = 1;                                   // tile_dim1 = 1, tile_dim2 = 0
        g1d[5] = Hdim;                                // tensor_dim0_stride[31:0] = 1024
        g1d[6] = 0;                                   // stride[47:32]=0 | tensor_dim1_stride lo
        g1d[7] = 0;                                   // tensor_dim1_stride hi

        v4i g2d;
        g2d[0] = 0;                                   // tensor_dim2 (unused, 2D tile)
        g2d[1] = ROWSTRIDE;                           // lds_addr_increment = 1032 elements
        g2d[2] = Hdim;                                // global_addr_increment[31:0] = 1024 elements
        g2d[3] = (TOK - 1) << 16;                     // incr[47:32]=0 | iterate_count = 15 (16x)

        v4i g3d = {0, 0, 0, 0};
        v8i g4d = {0, 0, 0, 0, 0, 0, 0, 0};

        __builtin_amdgcn_tensor_load_to_lds(g0, g1d, g2d, g3d, g4d, 0);
        __builtin_amdgcn_s_wait_tensorcnt(0);
    }
    __syncthreads();

    // ---------------- GEMM1: cat[16 x E] = x_tile @ Wc^T (per-wave K slice) ----------------
    v8f acc = {0.f, 0.f, 0.f, 0.f, 0.f, 0.f, 0.f, 0.f};
    {
        const int k0 = wave * (Hdim / NWAVE);
        const float* wrow = Wc + (size_t)eIdx * Hdim;
        for (int kk = 0; kk < Hdim / NWAVE; kk += 4) {
            int k = k0 + kk + koff;
            v2f a; a.x = xbuf[l16 * ROWSTRIDE + k];
                   a.y = xbuf[l16 * ROWSTRIDE + k + 1];
            v2f b; b.x = wrow[k];
                   b.y = wrow[k + 1];
            acc = __builtin_amdgcn_wmma_f32_16x16x4_f32(false, a, false, b,
                                                        (short)0, acc, false, false);
        }
    }
    for (int r = 0; r < 8; ++r) cscratch[wave][r + 8 * lhalf][l16] = acc[r];
    __syncthreads();

    // ---------------- Cusp dynamics (redundant across waves; wave0 publishes) ----------------
    {
        float bcv = bc[eIdx];
        float c0 = control[2 * eIdx + 0];
        float c1 = control[2 * eIdx + 1];
        float eqv[8];
        for (int r = 0; r < 8; ++r) {
            float s = 0.f;
            for (int w = 0; w < NWAVE; ++w) s += cscratch[w][r + 8 * lhalf][l16];
            float cat = s + bcv;
            float av = c0 + cat * 0.1f;
            float bv = c1 + cat * 0.05f;
            float e_ = 0.f;
            for (int it = 0; it < 10; ++it)
                e_ = e_ - 0.1f * (e_ * e_ * e_ + av * e_ + bv);
            eqv[r] = e_;
        }
        if (wave == 0 && l16 < Edim)
            for (int r = 0; r < 8; ++r) eqbuf[r + 8 * lhalf][l16] = eqv[r];
    }
    __syncthreads();

    // ---------------- GEMM2: pre1 = eq @ Wfc^T + bfc + x (in-place into xbuf) ----------------
    {
        v2f a0; a0.x = eqbuf[l16][koff];     a0.y = eqbuf[l16][koff + 1];
        v2f a1; a1.x = eqbuf[l16][4 + koff]; a1.y = eqbuf[l16][5 + koff];
        for (int t = 0; t < 16; ++t) {
            int nb = (wave * 16 + t) * 16;
            int hcol = nb + l16;
            float bfcv = bfc[hcol];
            v8f c = {bfcv, bfcv, bfcv, bfcv, bfcv, bfcv, bfcv, bfcv};
            const float* wp = Wfc + (size_t)hcol * Edim;
            v2f b0; b0.x = wp[koff];     b0.y = wp[koff + 1];
            v2f b1f; b1f.x = wp[4 + koff]; b1f.y = wp[5 + koff];
            c = __builtin_amdgcn_wmma_f32_16x16x4_f32(false, a0, false, b0, (short)0, c, false, false);
            c = __builtin_amdgcn_wmma_f32_16x16x4_f32(false, a1, false, b1f, (short)0, c, false, false);
            for (int r = 0; r < 8; ++r) {
                int idx = (r + 8 * lhalf) * ROWSTRIDE + hcol;
                xbuf[idx] = c[r] + xbuf[idx];    // add residual x, store pre-LN1
            }
        }
    }
    __syncthreads();

    // ---------------- LayerNorm-1 statistics ----------------
    {
        int token = tid >> 3, part = tid & 7;
        const float* row = &xbuf[token * ROWSTRIDE];
        float s = 0.f, sq = 0.f;
        for (int j = 0; j < 32; ++j) {
            v4f v = *(const v4f*)(row + (part + j * 8) * 4);
            s  += v.x + v.y + v.z + v.w;
            sq += v.x * v.x + v.y * v.y + v.z * v.z + v.w * v.w;
        }
        for (int d = 1; d < 8; d <<= 1) { s += __shfl_xor(s, d, 8); sq += __shfl_xor(sq, d, 8); }
        if (part == 0) {
            float m = s * (1.f / Hdim);
            float var = sq * (1.f / Hdim) - m * m;
            stats[token][0] = m;
            stats[token][1] = rsqrtf(var + 1e-5f);
        }
    }
    __syncthreads();

    // ---------------- GEMM3: substrate = relu(LN1(pre1) @ We^T + be) ----------------
    {
        float mtok = stats[l16][0], rtok = stats[l16][1];
        v8f acc3 = {0.f, 0.f, 0.f, 0.f, 0.f, 0.f, 0.f, 0.f};
        const int k0 = wave * (Hdim / NWAVE);
        const float* wrow = We + (size_t)eIdx * Hdim;
        for (int kk = 0; kk < Hdim / NWAVE; kk += 4) {
            int k = k0 + kk + koff;
            float v0 = xbuf[l16 * ROWSTRIDE + k];
            float v1 = xbuf[l16 * ROWSTRIDE + k + 1];
            v2f a; a.x = (v0 - mtok) * rtok * g1[k]     + b1[k];
                   a.y = (v1 - mtok) * rtok * g1[k + 1] + b1[k + 1];
            v2f b; b.x = wrow[k]; b.y = wrow[k + 1];
            acc3 = __builtin_amdgcn_wmma_f32_16x16x4_f32(false, a, false, b,
                                                         (short)0, acc3, false, false);
        }
        for (int r = 0; r < 8; ++r) cscratch[wave][r + 8 * lhalf][l16] = acc3[r];
    }
    __syncthreads();

    // ---------------- Fermentation dynamics ----------------
    {
        float bev = be[eIdx];
        float sub[8];
        for (int r = 0; r < 8; ++r) {
            float s = 0.f;
            for (int w = 0; w < NWAVE; ++w) s += cscratch[w][r + 8 * lhalf][l16];
            sub[r] = fmaxf(s + bev, 0.f);
        }
        if (wave == 0 && l16 < Edim)
            for (int r = 0; r < 8; ++r) subbuf[r + 8 * lhalf][l16] = sub[r];
        __syncthreads();

        float sigW[8];
        for (int e2 = 0; e2 < 8; ++e2) {
            float w = Winh[e2 * Edim + eIdx];
            sigW[e2] = 1.f / (1.f + __expf(-w));
        }
        float Kmv = Km[eIdx], Vmv = Vmax[eIdx], gr = growth_rate_p[0];
        float prod[8];
        for (int r = 0; r < 8; ++r) {
            int token = r + 8 * lhalf;
            float inh = 0.f;
            for (int e2 = 0; e2 < 8; ++e2) inh += subbuf[token][e2] * sigW[e2];
            float Km_app = Kmv * (1.f + inh);
            float rate = Vmv * sub[r] / (Km_app + sub[r] + 1e-10f);
            float s = rate;
            for (int d = 1; d < 8; d <<= 1) s += __shfl_xor(s, d, 8);  // mean over 8 enzymes
            float mrate = s * (1.f / Edim);
            float growth = 0.5f * mrate / (0.1f + mrate + 1e-10f);
            prod[r] = rate * (1.f + growth * gr);
        }
        if (wave == 0 && l16 < Edim)
            for (int r = 0; r < 8; ++r) prodbuf[r + 8 * lhalf][l16] = prod[r];
    }
    __syncthreads();

    // ---------------- GEMM4: pre2 = product @ Wo^T + bo + LN1(pre1) (in place) ----------------
    {
        v2f a0; a0.x = prodbuf[l16][koff];     a0.y = prodbuf[l16][koff + 1];
        v2f a1; a1.x = prodbuf[l16][4 + koff]; a1.y = prodbuf[l16][5 + koff];
        for (int t = 0; t < 16; ++t) {
            int nb = (wave * 16 + t) * 16;
            int hcol = nb + l16;
            float bov = bo[hcol];
            float g1v = g1[hcol], b1v = b1[hcol];
            v8f c = {bov, bov, bov, bov, bov, bov, bov, bov};
            const float* wp = Wo + (size_t)hcol * Edim;
            v2f b0; b0.x = wp[koff];     b0.y = wp[koff + 1];
            v2f b1f; b1f.x = wp[4 + koff]; b1f.y = wp[5 + koff];
            c = __builtin_amdgcn_wmma_f32_16x16x4_f32(false, a0, false, b0, (short)0, c, false, false);
            c = __builtin_amdgcn_wmma_f32_16x16x4_f32(false, a1, false, b1f, (short)0, c, false, false);
            for (int r = 0; r < 8; ++r) {
                int token = r + 8 * lhalf;
                int idx = token * ROWSTRIDE + hcol;
                float hv = (xbuf[idx] - stats[token][0]) * stats[token][1] * g1v + b1v;
                xbuf[idx] = c[r] + hv;   // store pre-LN2
            }
        }
    }
    __syncthreads();

    // ---------------- LayerNorm-2 statistics ----------------
    {
        int token = tid >> 3, part = tid & 7;
        const float* row = &xbuf[token * ROWSTRIDE];
        float s = 0.f, sq = 0.f;
        for (int j = 0; j < 32; ++j) {
            v4f v = *(const v4f*)(row + (part + j * 8) * 4);
            s  += v.x + v.y + v.z + v.w;
            sq += v.x * v.x + v.y * v.y + v.z * v.z + v.w * v.w;
        }
        for (int d = 1; d < 8; d <<= 1) { s += __shfl_xor(s, d, 8); sq += __shfl_xor(sq, d, 8); }
        if (part == 0) {
            float m = s * (1.f / Hdim);
            float var = sq * (1.f / Hdim) - m * m;
            stats[token][2] = m;
            stats[token][3] = rsqrtf(var + 1e-5f);
        }
    }
    __syncthreads();

    // ---------------- Output: normalize with g2/b2, coalesced store ----------------
    {
        float* og = out + (size_t)tokbase * Hdim;
        for (int j = 0; j < 32; ++j) {
            int idx = tid * 4 + j * 512;
            int row = idx >> 10;
            int col = idx & (Hdim - 1);
            v4f v = *(const v4f*)(&xbuf[row * ROWSTRIDE + col]);
            float m = stats[row][2], rs = stats[row][3];
            v4f g = *(const v4f*)(g2 + col);
            v4f bb = *(const v4f*)(b2 + col);
            v4f o;
            o.x = (v.x - m) * rs * g.x + bb.x;
            o.y = (v.y - m) * rs * g.y + bb.y;
            o.z = (v.z - m) * rs * g.z + bb.z;
            o.w = (v.w - m) * rs * g.w + bb.w;
            *(v4f*)(og + idx) = o;
        }
    }
}

extern "C" void kernel_launch(void* const* d_in, const int* in_sizes, int n_in,
                              void* d_out, int out_size, void* d_ws, size_t ws_size,
                              hipStream_t stream) {
    const float* x           = (const float*)d_in[0];
    const float* control     = (const float*)d_in[1];
    const float* Wc          = (const float*)d_in[2];
    const float* bc          = (const float*)d_in[3];
    const float* Wfc         = (const float*)d_in[4];
    const float* bfc         = (const float*)d_in[5];
    const float* g1          = (const float*)d_in[6];
    const float* b1          = (const float*)d_in[7];
    const float* We          = (const float*)d_in[8];
    const float* be          = (const float*)d_in[9];
    const float* Vmax        = (const float*)d_in[10];
    const float* Km          = (const float*)d_in[11];
    const float* Winh        = (const float*)d_in[12];
    const float* growth_rate = (const float*)d_in[13];
    const float* Wo          = (const float*)d_in[14];
    const float* bo          = (const float*)d_in[15];
    const float* g2          = (const float*)d_in[16];
    const float* b2          = (const float*)d_in[17];
    float* out = (float*)d_out;

    const int n_tokens = 8 * 2048;          // B * S
    dim3 grid(n_tokens / TOK);              // 1024 workgroups
    dim3 block(128);                        // 4 waves (wave32)
    fused_catastrophe_fermentation<<<grid, block, 0, stream>>>(
        x, control, Wc, bc, Wfc, bfc, g1, b1,
        We, be, Vmax, Km, Winh, growth_rate, Wo, bo, g2, b2, out);
}